// Cov_2_10806137716744
// MI455X (gfx1250) — compile-verified
//
#include <hip/hip_runtime.h>

// Problem constants (from reference): N=8192 rows, D=512 features.
static constexpr int kN = 8192;
static constexpr int kD = 512;

static constexpr int TILE_I = 64;    // workgroup tile rows (i)
static constexpr int TILE_J = 256;   // workgroup tile cols (j)
static constexpr int KC     = 32;    // K chunk staged in LDS
static constexpr int LDSS   = 36;    // padded LDS row stride (floats) -> conflict-free ds_load_b64
static constexpr int NCHUNK = kD / KC;     // 16 (even -> 2x-unrolled pipeline below)
// staging: (TILE_I + TILE_J) rows * (KC/4) b128 = 2560 / 256 threads = 10 per thread
static constexpr int NSTG   = (TILE_I + TILE_J) * (KC / 4) / 256;

typedef float v2f __attribute__((ext_vector_type(2)));
typedef float v8f __attribute__((ext_vector_type(8)));

// Wait until this wave's outstanding async (global->LDS) ops <= n (compile-time literal).
#if defined(__has_builtin) && __has_builtin(__builtin_amdgcn_s_wait_asynccnt)
#define WAIT_ASYNCCNT(n) __builtin_amdgcn_s_wait_asynccnt(n)
#else
#define WAIT_ASYNCCNT(n) asm volatile("s_wait_asynccnt " #n ::: "memory")
#endif

// One b128 async copy: global (64-bit vaddr, GV mode) -> LDS (32-bit lds addr).
// No VGPR data registers are used; tracked by ASYNCcnt.
__device__ __forceinline__ void async_b128_to_lds(uint32_t lds_addr, uint64_t gaddr) {
    asm volatile("global_load_async_to_lds_b128 %0, %1, off"
                 :: "v"(lds_addr), "v"(gaddr)
                 : "memory");
}

// Issue one chunk of async copies into buffer BUF (compile-time!), then advance
// the per-slot global pointers by KC floats.  la[] holds buffer-0 LDS byte
// addresses; buffer-1 addresses differ by a compile-time constant delta.
template <int BUF>
__device__ __forceinline__ void issue_chunk(uint64_t (&ga)[NSTG], const uint32_t (&la)[NSTG]) {
    #pragma unroll
    for (int it = 0; it < NSTG; ++it) {
        const uint32_t delta = (it < 2) ? (uint32_t)(TILE_I * LDSS * sizeof(float))
                                        : (uint32_t)(TILE_J * LDSS * sizeof(float));
        const uint32_t lds = la[it] + (BUF ? delta : 0u);
        async_b128_to_lds(lds, ga[it]);
        ga[it] += KC * sizeof(float);
    }
}

// 8 k-steps x 8 fp32 WMMAs on one staged K-chunk (wave tile 32x64 = 2x4 accumulators).
__device__ __forceinline__ void mm_chunk(const float* __restrict__ Asl,
                                         const float* __restrict__ Bsl,
                                         v8f (&acc)[2][4],
                                         int wr, int wc, int l16, int koff) {
    #pragma unroll
    for (int kk = 0; kk < KC; kk += 4) {
        v2f a0 = *(const v2f*)(&Asl[(wr * 32 +      l16) * LDSS + kk + koff]);
        v2f a1 = *(const v2f*)(&Asl[(wr * 32 + 16 + l16) * LDSS + kk + koff]);
        v2f b0 = *(const v2f*)(&Bsl[(wc * 64 +      l16) * LDSS + kk + koff]);
        v2f b1 = *(const v2f*)(&Bsl[(wc * 64 + 16 + l16) * LDSS + kk + koff]);
        v2f b2 = *(const v2f*)(&Bsl[(wc * 64 + 32 + l16) * LDSS + kk + koff]);
        v2f b3 = *(const v2f*)(&Bsl[(wc * 64 + 48 + l16) * LDSS + kk + koff]);
        acc[0][0] = __builtin_amdgcn_wmma_f32_16x16x4_f32(false, a0, false, b0,
                                                          (short)0, acc[0][0], false, false);
        acc[0][1] = __builtin_amdgcn_wmma_f32_16x16x4_f32(false, a0, false, b1,
                                                          (short)0, acc[0][1], false, false);
        acc[0][2] = __builtin_amdgcn_wmma_f32_16x16x4_f32(false, a0, false, b2,
                                                          (short)0, acc[0][2], false, false);
        acc[0][3] = __builtin_amdgcn_wmma_f32_16x16x4_f32(false, a0, false, b3,
                                                          (short)0, acc[0][3], false, false);
        acc[1][0] = __builtin_amdgcn_wmma_f32_16x16x4_f32(false, a1, false, b0,
                                                          (short)0, acc[1][0], false, false);
        acc[1][1] = __builtin_amdgcn_wmma_f32_16x16x4_f32(false, a1, false, b1,
                                                          (short)0, acc[1][1], false, false);
        acc[1][2] = __builtin_amdgcn_wmma_f32_16x16x4_f32(false, a1, false, b2,
                                                          (short)0, acc[1][2], false, false);
        acc[1][3] = __builtin_amdgcn_wmma_f32_16x16x4_f32(false, a1, false, b3,
                                                          (short)0, acc[1][3], false, false);
    }
}

// ---------------------------------------------------------------------------
// Row squared norms: n[i] = sum_k seq[i][k]^2.  One wave32 per row.
// ---------------------------------------------------------------------------
__global__ __launch_bounds__(256) void norms_kernel(const float* __restrict__ seq,
                                                    float* __restrict__ nrm) {
    const int wave = threadIdx.x >> 5;
    const int lane = threadIdx.x & 31;
    const int row  = blockIdx.x * 8 + wave;
    const float* p = seq + (size_t)row * kD;
    float s = 0.f;
    #pragma unroll
    for (int k = lane; k < kD; k += 32) {
        float v = p[k];
        s += v * v;
    }
    #pragma unroll
    for (int off = 16; off > 0; off >>= 1) s += __shfl_xor(s, off, 32);
    if (lane == 0) nrm[row] = s;
}

// ---------------------------------------------------------------------------
// Zero the accumulators (rowsum[kN], total).
// ---------------------------------------------------------------------------
__global__ void init_kernel(float* __restrict__ rowsum, float* __restrict__ total) {
    const int i = blockIdx.x * blockDim.x + threadIdx.x;
    if (i < kN) rowsum[i] = 0.f;
    if (i == 0) *total = 0.f;
}

// ---------------------------------------------------------------------------
// count = sum(sum_idx != 0)
// ---------------------------------------------------------------------------
__global__ __launch_bounds__(256) void count_kernel(const float* __restrict__ smk,
                                                    float* __restrict__ countOut) {
    __shared__ float sdata[256];
    float s = 0.f;
    for (int i = threadIdx.x; i < kN; i += 256) s += (smk[i] != 0.f) ? 1.f : 0.f;
    sdata[threadIdx.x] = s;
    __syncthreads();
    for (int off = 128; off > 0; off >>= 1) {
        if (threadIdx.x < (unsigned)off) sdata[threadIdx.x] += sdata[threadIdx.x + off];
        __syncthreads();
    }
    if (threadIdx.x == 0) *countOut = sdata[0];
}

// ---------------------------------------------------------------------------
// Main fused kernel: fp32 WMMA Gram tile (seq . seq^T), mask/norm dist
// epilogue, fused masked-row-sum and total-sum reductions.
//
// Block: 256 threads = 8 waves arranged 2 (i) x 4 (j).
// Wave tile: 32 x 64 -> 2x4 accumulators of v_wmma_f32_16x16x4_f32
// (inner step: 2 A + 4 B fragment loads feed 8 WMMAs).
//
// Staging uses GLOBAL_LOAD_ASYNC_TO_LDS_B128 with double-buffered LDS and a
// 2x-unrolled pipeline so the buffer selector is a compile-time constant
// (no v_movrels register indexing in the prefetch phase).  ASYNCcnt is
// in-order, so s_wait_asynccnt <= NSTG means the current chunk has landed.
//
// f32 16x16x4 fragment layout (ISA 7.12.2):
//   A (16x4, MxK): lanes 0-15 -> M=lane,    v0:K=0, v1:K=1
//                  lanes16-31 -> M=lane-16, v0:K=2, v1:K=3
//   B (4x16, KxN): mirrored (N=lane&15, K offset 2*(lane>>4))
//   C (16x16):     N=lane&15, VGPR v -> M = v + 8*(lane>>4)
// ---------------------------------------------------------------------------
__global__ __launch_bounds__(256) void dist_kernel(const float* __restrict__ seq,
                                                   const float* __restrict__ qvs,
                                                   const float* __restrict__ smk,
                                                   const float* __restrict__ nrm,
                                                   float* __restrict__ rowsum,
                                                   float* __restrict__ total) {
    __shared__ float As[2][TILE_I * LDSS];
    __shared__ float Bs[2][TILE_J * LDSS];
    __shared__ float rowacc[TILE_I];
    __shared__ float blkTotal;

    const int tid  = threadIdx.x;
    const int wave = tid >> 5;
    const int lane = tid & 31;
    const int half = lane >> 4;   // 0: lanes 0-15, 1: lanes 16-31
    const int l16  = lane & 15;
    const int wr   = wave >> 2;   // wave row 0..1  (i, 32 rows each)
    const int wc   = wave & 3;    // wave col 0..3  (j, 64 cols each)
    const int koff = 2 * half;    // K sub-offset of this lane's fragment

    const int i0 = blockIdx.y * TILE_I;
    const int j0 = blockIdx.x * TILE_J;

    if (tid < TILE_I) rowacc[tid] = 0.f;
    if (tid == 0)     blkTotal = 0.f;

    // Per-thread staging slots (it < 2 -> A rows 0..63, else B rows 0..255).
    // idx = tid + it*256; row = idx>>3 in [0,320); c4 = (idx&7)*4 floats.
    uint64_t ga[NSTG];   // running global byte address of this slot
    uint32_t la[NSTG];   // LDS byte address of this slot in buffer 0
    #pragma unroll
    for (int it = 0; it < NSTG; ++it) {
        const int idx  = tid + it * 256;
        const int row  = idx >> 3;
        const int c4   = (idx & 7) << 2;
        const int grow = (it < 2) ? (i0 + row) : (j0 + row - TILE_I);
        ga[it] = (uint64_t)(uintptr_t)(seq + (size_t)grow * kD + c4);
        float* dst = (it < 2) ? &As[0][row * LDSS + c4]
                              : &Bs[0][(row - TILE_I) * LDSS + c4];
        la[it] = (uint32_t)(uintptr_t)dst;
    }

    v8f acc[2][4] = {};

    // Prologue: chunk 0 -> buffer 0.
    issue_chunk<0>(ga, la);

    // 2x-unrolled double-buffered pipeline (NCHUNK is even).
    for (int c = 0; c < NCHUNK; c += 2) {
        // --- even chunk: prefetch -> buf1, compute buf0 ---
        issue_chunk<1>(ga, la);          // c+1 < NCHUNK always (c <= NCHUNK-2)
        WAIT_ASYNCCNT(NSTG);             // chunk c landed (in-order completion)
        __syncthreads();
        mm_chunk(As[0], Bs[0], acc, wr, wc, l16, koff);
        __syncthreads();                 // buf0 consumed

        // --- odd chunk: prefetch -> buf0, compute buf1 ---
        if (c + 2 < NCHUNK) {
            issue_chunk<0>(ga, la);
            WAIT_ASYNCCNT(NSTG);
        } else {
            WAIT_ASYNCCNT(0);
        }
        __syncthreads();
        mm_chunk(As[1], Bs[1], acc, wr, wc, l16, koff);
        __syncthreads();                 // buf1 consumed
    }

    // ---- Epilogue: masks/norms -> dist, fused reductions ----
    // Uniform formula valid for all mask cases (masks are 0/1):
    //   d2 = qm_i*n_i + sm_j*n_j - 2*qm_i*sm_j*g_ij
    float qmL[2][8], niL[2][8];
    #pragma unroll
    for (int si = 0; si < 2; ++si) {
        #pragma unroll
        for (int v = 0; v < 8; ++v) {
            const int gi = i0 + wr * 32 + si * 16 + v + 8 * half;
            qmL[si][v] = qvs[gi];
            niL[si][v] = nrm[gi];
        }
    }

    float totLocal = 0.f;
    #pragma unroll
    for (int sj = 0; sj < 4; ++sj) {
        const int gj   = j0 + wc * 64 + sj * 16 + l16;  // this lane's column
        const float sm = smk[gj];
        const float nj = nrm[gj];
        #pragma unroll
        for (int si = 0; si < 2; ++si) {
            #pragma unroll
            for (int v = 0; v < 8; ++v) {
                const float qm = qmL[si][v];
                const float ni = niL[si][v];
                const float g  = acc[si][sj][v];
                float d2 = qm * ni + sm * nj - 2.f * qm * sm * g;
                d2 = fmaxf(d2, 0.f);
                const float dist = (d2 > 0.f) ? sqrtf(d2) : 0.f;
                totLocal += dist;
                // masked row sum over the 16 columns held by this half-wave
                float m = dist * sm;
                m += __shfl_xor(m, 8, 16);
                m += __shfl_xor(m, 4, 16);
                m += __shfl_xor(m, 2, 16);
                m += __shfl_xor(m, 1, 16);
                if (l16 == 0)
                    atomicAdd(&rowacc[wr * 32 + si * 16 + v + 8 * half], m);
            }
        }
    }
    atomicAdd(&blkTotal, totLocal);
    __syncthreads();

    if (tid < TILE_I) atomicAdd(&rowsum[i0 + tid], rowacc[tid]);
    if (tid == 0)     atomicAdd(total, blkTotal);
}

// ---------------------------------------------------------------------------
// out[i] = (1 - min(rowsum[i]/count, norm)/norm) * w + b,  norm = total/N^2
// ---------------------------------------------------------------------------
__global__ void finalize_kernel(const float* __restrict__ rowsum,
                                const float* __restrict__ total,
                                const float* __restrict__ count,
                                const float* __restrict__ weight,
                                const float* __restrict__ bias,
                                float* __restrict__ out) {
    const int i = blockIdx.x * blockDim.x + threadIdx.x;
    if (i >= kN) return;
    const float norm = *total / ((float)kN * (float)kN);
    float r = rowsum[i] / *count;
    r = fminf(r, norm);
    const float s = 1.f - r / norm;
    out[i] = s * weight[0] + bias[0];
}

// ---------------------------------------------------------------------------
extern "C" void kernel_launch(void* const* d_in, const int* in_sizes, int n_in,
                              void* d_out, int out_size, void* d_ws, size_t ws_size,
                              hipStream_t stream) {
    const float* seq  = (const float*)d_in[0];   // [N, D]
    const float* qvs  = (const float*)d_in[1];   // [N, 1] in {0,1}
    const float* smk  = (const float*)d_in[2];   // [N, 1] in {0,1}
    const float* wgt  = (const float*)d_in[3];   // [1, 1]
    const float* bias = (const float*)d_in[4];   // [1]
    float* out = (float*)d_out;                  // [N, 1]

    float* ws     = (float*)d_ws;
    float* rowsum = ws;              // kN floats
    float* nrm    = ws + kN;         // kN floats
    float* total  = ws + 2 * kN;     // 1 float
    float* count  = ws + 2 * kN + 1; // 1 float

    init_kernel<<<(kN + 255) / 256, 256, 0, stream>>>(rowsum, total);
    count_kernel<<<1, 256, 0, stream>>>(smk, count);
    norms_kernel<<<kN / 8, 256, 0, stream>>>(seq, nrm);

    dim3 grid(kN / TILE_J, kN / TILE_I);   // 32 x 128 = 4096 workgroups
    dist_kernel<<<grid, 256, 0, stream>>>(seq, qvs, smk, nrm, rowsum, total);

    finalize_kernel<<<(kN + 255) / 256, 256, 0, stream>>>(rowsum, total, count, wgt, bias, out);
}